// Rendering_27599459844269
// MI455X (gfx1250) — compile-verified
//
#include <hip/hip_runtime.h>

// CDNA5 / gfx1250 volume-rendering composite (NeRF ray compositing).
// Exclusive cumprod of transmittance done in log space as a two-level scan on
// the f32 matrix pipe: V_WMMA_F32_16X16X4_F32 with (a) tile-invariant
// strict-upper-triangular B fragments for the intra-tile exclusive cumsum and
// (b) an all-ones B accumulating the inter-tile carry. Fully f32-exact GEMM.

typedef float v2f __attribute__((ext_vector_type(2)));
typedef float v8f __attribute__((ext_vector_type(8)));

#define N_SAMPLES      128
#define RAYS_PER_BLOCK 16
#define EPS_T          1e-10f
#define LAST_DIST      1e10f

// LDS layouts (strides coprime with 64 banks -> conflict-free lane patterns)
#define T_STRIDE   17   // t_lds[s*17 + r]
#define RED_STRIDE 25   // red_lds[lane*25 + v*3 + c]

__global__ __launch_bounds__(32)
void Rendering_27599459844269_kernel(const float* __restrict__ raw,
                                     const float* __restrict__ z_vals,
                                     float* __restrict__ out)
{
    __shared__ float t_lds[N_SAMPLES * T_STRIDE];   // 8704 B
    __shared__ float red_lds[32 * RED_STRIDE];      // 3200 B

    const int lane = threadIdx.x;        // 0..31 (wave32)
    const int ray0 = blockIdx.x * RAYS_PER_BLOCK;

    // ---------------- Phase A: stream density+z once, fill t_lds ----------------
    // element e -> ray r = e>>7, sample s = e&127. Lanes walk consecutive
    // samples; the stride-16B density loads still touch each raw cacheline
    // exactly once, so HBM traffic is unchanged vs float4 loads.
    for (int e = lane; e < RAYS_PER_BLOCK * N_SAMPLES; e += 32) {
        const int r = e >> 7;
        const int s = e & (N_SAMPLES - 1);
        const size_t base = ((size_t)(ray0 + r) * N_SAMPLES + s);

        const float sigma = raw[base * 4 + 3];
        const float z0    = z_vals[base];
        const float dist  = (s < N_SAMPLES - 1) ? (z_vals[base + 1] - z0) : LAST_DIST;

        const float x = fmaxf(sigma, 0.0f) * dist;  // relu(sigma)*dist
        const float y = __expf(-x);                 // 1 - alpha
        // t = log(1 - alpha + eps); cumsum(t) == log of exclusive cumprod
        t_lds[s * T_STRIDE + r] = __logf(y + EPS_T);
    }
    __syncthreads();

    // ---------------- Phase B: two-level WMMA scan + fused weights/colors ----------------
    // A (16x4 f32): lane L holds row m=L&15; VGPR0 -> K=2h, VGPR1 -> K=2h+1 (h=L>>4)
    // B (4x16 f32): VGPR0 -> rows {0,2} (half-sel), VGPR1 -> rows {1,3}; col = L&15
    // C/D (16x16):  VGPR v, half 0 -> M=v, half 1 -> M=v+8; N = L&15
    const int h  = lane >> 4;
    const int nl = lane & 15;

    // Tile-invariant strict-upper B fragments (B[j,i] = j<i) + all-ones B.
    v2f bs[4];
#pragma unroll
    for (int kk = 0; kk < 4; ++kk) {
        bs[kk].x = (kk * 4 + 2 * h     < nl) ? 1.0f : 0.0f;
        bs[kk].y = (kk * 4 + 2 * h + 1 < nl) ? 1.0f : 0.0f;
    }
    const v2f bones = {1.0f, 1.0f};

    v8f carry = {0.0f, 0.0f, 0.0f, 0.0f, 0.0f, 0.0f, 0.0f, 0.0f};
    float partial[8][3];
#pragma unroll
    for (int v = 0; v < 8; ++v) {
        partial[v][0] = 0.0f; partial[v][1] = 0.0f; partial[v][2] = 0.0f;
    }

    for (int nt = 0; nt < N_SAMPLES / 16; ++nt) {
        const int n0 = nt * 16;
        const int n  = n0 + nl;          // this lane's output sample

        // Load this tile's A fragments once (compiler fuses to ds_load_2addr).
        v2f a[4];
#pragma unroll
        for (int kk = 0; kk < 4; ++kk) {
            const int ka = n0 + kk * 4 + 2 * h;
            a[kk].x = t_lds[(ka    ) * T_STRIDE + nl];
            a[kk].y = t_lds[(ka + 1) * T_STRIDE + nl];
        }

        // exclusive cumsum = inter-tile carry + intra-tile strict-upper part
        v8f acc = carry;
#pragma unroll
        for (int kk = 0; kk < 4; ++kk)
            acc = __builtin_amdgcn_wmma_f32_16x16x4_f32(
                      false, a[kk], false, bs[kk], (short)0, acc, false, false);

        if (nt != N_SAMPLES / 16 - 1) {
#pragma unroll
            for (int kk = 0; kk < 4; ++kk)
                carry = __builtin_amdgcn_wmma_f32_16x16x4_f32(
                            false, a[kk], false, bones, (short)0, carry, false, false);
        }

        // Fuse alpha/weight/color. raw re-read is contiguous 256B per half-wave
        // and L0/L2 resident (192MB L2), so no extra HBM traffic.
#pragma unroll
        for (int v = 0; v < 8; ++v) {
            const int m = v + 8 * h;
            const size_t base = ((size_t)(ray0 + m) * N_SAMPLES + n);
            const float4 rv = *(const float4*)(raw + base * 4);

            const float tval  = t_lds[n * T_STRIDE + m];
            const float alpha = 1.0f + EPS_T - __expf(tval);   // 1 - (y+eps) + eps
            const float w     = alpha * __expf(acc[v]);        // alpha * trans_excl

            partial[v][0] += w / (1.0f + __expf(-rv.x));
            partial[v][1] += w / (1.0f + __expf(-rv.y));
            partial[v][2] += w / (1.0f + __expf(-rv.z));
        }
    }

    // ---------------- Phase C: cross-lane reduction (16 lanes per ray-half) ----------------
#pragma unroll
    for (int v = 0; v < 8; ++v) {
        red_lds[lane * RED_STRIDE + v * 3 + 0] = partial[v][0];
        red_lds[lane * RED_STRIDE + v * 3 + 1] = partial[v][1];
        red_lds[lane * RED_STRIDE + v * 3 + 2] = partial[v][2];
    }
    __syncthreads();

    for (int o = lane; o < RAYS_PER_BLOCK * 3; o += 32) {
        const int m  = o / 3;
        const int c  = o - m * 3;
        const int hh = m >> 3;
        const int vv = m & 7;
        float s = 0.0f;
#pragma unroll
        for (int i = 0; i < 16; ++i)
            s += red_lds[(hh * 16 + i) * RED_STRIDE + vv * 3 + c];
        out[(size_t)(ray0 + m) * 3 + c] = s;
    }
}

extern "C" void kernel_launch(void* const* d_in, const int* in_sizes, int n_in,
                              void* d_out, int out_size, void* d_ws, size_t ws_size,
                              hipStream_t stream)
{
    const float* raw    = (const float*)d_in[0];
    const float* z_vals = (const float*)d_in[1];
    float* out = (float*)d_out;

    const int n_rays = in_sizes[1] / N_SAMPLES;           // 65536
    const int blocks = n_rays / RAYS_PER_BLOCK;           // 4096 waves
    Rendering_27599459844269_kernel<<<blocks, 32, 0, stream>>>(raw, z_vals, out);
}